// Predictor_14757507629579
// MI455X (gfx1250) — compile-verified
//
#include <hip/hip_runtime.h>

typedef float v2f __attribute__((ext_vector_type(2)));
typedef float v8f __attribute__((ext_vector_type(8)));

#define BATCH 16384
#define NSTEP 26
#define OUTSTRIDE 1600  /* 25*64 */

__device__ __forceinline__ float sigf(float x) { return 1.0f / (1.0f + __expf(-x)); }

// ---------------- K0: zero global stat accumulators (replayed each launch) ---
__global__ __launch_bounds__(256) void k_zero(float* gstats) {
  gstats[threadIdx.x] = 0.0f;
}

// ---------------- K1: a1 = concat(x0,x1,x2) @ W1.T + b1 ; accumulate stats ---
// 8 rows per WG, one output element per thread (j = t&31, r = t>>5).
__global__ __launch_bounds__(256) void k_mlp1(const float* __restrict__ x0,
                                              const float* __restrict__ x1,
                                              const float* __restrict__ x2,
                                              const float* __restrict__ W1,
                                              const float* __restrict__ b1,
                                              float* __restrict__ a1,
                                              float* __restrict__ gstats) {
  __shared__ float W1s[384 * 32];  // transposed [k][j], 48KB
  __shared__ float xs[8 * 384];    // 12KB
  __shared__ float ssum[32], ssq[32];
  const int t = threadIdx.x;
  const int row0 = blockIdx.x * 8;

  for (int idx = t; idx < 384 * 32; idx += 256) {
    int j = idx & 31, k = idx >> 5;
    W1s[k * 32 + j] = W1[j * 384 + k];
  }
  for (int idx = t; idx < 8 * 384; idx += 256) {
    int r = idx / 384, k = idx - r * 384;
    float v;
    if (k < 128)       v = x0[(row0 + r) * 128 + k];
    else if (k < 256)  v = x1[(row0 + r) * 128 + (k - 128)];
    else               v = x2[(row0 + r) * 128 + (k - 256)];
    xs[idx] = v;
  }
  if (t < 32) { ssum[t] = 0.0f; ssq[t] = 0.0f; }
  __syncthreads();

  const int j = t & 31, r = t >> 5;
  float acc = b1[j];
  for (int k = 0; k < 384; ++k) acc += xs[r * 384 + k] * W1s[k * 32 + j];
  a1[(row0 + r) * 32 + j] = acc;
  atomicAdd(&ssum[j], acc);
  atomicAdd(&ssq[j], acc * acc);
  __syncthreads();
  if (t < 32) {
    atomicAdd(&gstats[t], ssum[t]);
    atomicAdd(&gstats[32 + t], ssq[t]);
  }
}

// ---------------- K2: h1 = relu(bn1(a1)); a2 = h1 @ W2.T + b2 ; stats2 -------
__global__ __launch_bounds__(256) void k_mlp2(const float* __restrict__ a1,
                                              const float* __restrict__ W2,
                                              const float* __restrict__ b2,
                                              const float* __restrict__ g1,
                                              const float* __restrict__ beta1,
                                              float* __restrict__ a2,
                                              float* __restrict__ gstats) {
  __shared__ float W2s[32 * 32];  // transposed [k][j]
  __shared__ float sc[32], sh[32];
  __shared__ float ssum[32], ssq[32];
  const int t = threadIdx.x;
  for (int idx = t; idx < 1024; idx += 256) {
    int j = idx & 31, k = idx >> 5;
    W2s[k * 32 + j] = W2[j * 32 + k];
  }
  if (t < 32) {
    float mean = gstats[t] * (1.0f / (float)BATCH);
    float var  = gstats[32 + t] * (1.0f / (float)BATCH) - mean * mean;
    float s = g1[t] * rsqrtf(var + 1e-5f);
    sc[t] = s;
    sh[t] = beta1[t] - mean * s;
    ssum[t] = 0.0f; ssq[t] = 0.0f;
  }
  __syncthreads();

  const int row = blockIdx.x * 256 + t;
  float h[32];
#pragma unroll
  for (int k = 0; k < 32; ++k) {
    float v = a1[row * 32 + k] * sc[k] + sh[k];
    h[k] = v > 0.0f ? v : 0.0f;
  }
  for (int jj = 0; jj < 32; ++jj) {
    float acc = b2[jj];
#pragma unroll
    for (int k = 0; k < 32; ++k) acc += h[k] * W2s[k * 32 + jj];
    a2[row * 32 + jj] = acc;
    atomicAdd(&ssum[jj], acc);
    atomicAdd(&ssq[jj], acc * acc);
  }
  __syncthreads();
  if (t < 32) {
    atomicAdd(&gstats[64 + t], ssum[t]);
    atomicAdd(&gstats[96 + t], ssq[t]);
  }
}

// ---------------- K3: h2 = relu(bn2(a2)); m = h2 @ W3.T + b3 -----------------
__global__ __launch_bounds__(256) void k_mlp3(const float* __restrict__ a2,
                                              const float* __restrict__ W3,
                                              const float* __restrict__ b3,
                                              const float* __restrict__ g2,
                                              const float* __restrict__ beta2,
                                              const float* __restrict__ gstats,
                                              float* __restrict__ mbuf) {
  __shared__ float W3s[32 * 64];  // transposed [k][j]
  __shared__ float sc[32], sh[32];
  const int t = threadIdx.x;
  for (int idx = t; idx < 2048; idx += 256) {
    int j = idx & 63, k = idx >> 6;
    W3s[k * 64 + j] = W3[j * 32 + k];
  }
  if (t < 32) {
    float mean = gstats[64 + t] * (1.0f / (float)BATCH);
    float var  = gstats[96 + t] * (1.0f / (float)BATCH) - mean * mean;
    float s = g2[t] * rsqrtf(var + 1e-5f);
    sc[t] = s;
    sh[t] = beta2[t] - mean * s;
  }
  __syncthreads();

  const int row = blockIdx.x * 256 + t;
  float h[32];
#pragma unroll
  for (int k = 0; k < 32; ++k) {
    float v = a2[row * 32 + k] * sc[k] + sh[k];
    h[k] = v > 0.0f ? v : 0.0f;
  }
  for (int jj = 0; jj < 64; ++jj) {
    float acc = b3[jj];
#pragma unroll
    for (int k = 0; k < 32; ++k) acc += h[k] * W3s[k * 64 + jj];
    mbuf[row * 64 + jj] = acc;
  }
}

// ---------------- K4: persistent-tile LSTM, fp32 WMMA 16x16x4 ----------------
// 64 batch rows per WG, 256 threads = 8 waves. Steps >=1 use merged Wih+Whh.
__global__ __launch_bounds__(256) void k_lstm(const float* __restrict__ mbuf,
                                              const float* __restrict__ Wih,
                                              const float* __restrict__ Whh,
                                              const float* __restrict__ bih,
                                              const float* __restrict__ bhh,
                                              float* __restrict__ out) {
  __shared__ v2f   wb[32 * 256];   // 64KB: wb[k/2][n] = (W.T[k][n], W.T[k+1][n])
  __shared__ float gb[64 * 256];   // 64KB gate buffer
  __shared__ float hb[64 * 64];    // 16KB current input/hidden
  __shared__ float bs[256];        // bias sum
  const int t = threadIdx.x;
  const int row0 = blockIdx.x * 64;

  // prologue: interleave Wih.T into wb; load m into hb; bias sum
  for (int idx = t; idx < 32 * 256; idx += 256) {
    int kp = idx >> 8, n = idx & 255;
    v2f v;
    v.x = Wih[n * 64 + 2 * kp];
    v.y = Wih[n * 64 + 2 * kp + 1];
    wb[idx] = v;
  }
  for (int idx = t; idx < 64 * 64; idx += 256) {
    int r = idx >> 6, k = idx & 63;
    hb[idx] = mbuf[(size_t)(row0 + r) * 64 + k];
  }
  bs[t] = bih[t] + bhh[t];

  // elementwise mapping: thread -> (row er, 16 cols starting at ec)
  const int er = t >> 2;
  const int ec = (t & 3) * 16;
  float c[16];
#pragma unroll
  for (int i = 0; i < 16; ++i) c[i] = 0.0f;

  // WMMA mapping: wave w -> M-tile (w>>1), N-tiles (w&1)*8 .. +7
  const int w  = t >> 5;
  const int l  = t & 31;
  const int mt = w >> 1;
  const int ntBase = (w & 1) * 8;
  const int lh = l >> 4;   // lane half selects K-pair
  const int ll = l & 15;

  __syncthreads();

  for (int s = 0; s < NSTEP; ++s) {
    // ---- WMMA phase: gates(64x256) = hb(64x64) @ W.T, K=64 in 16 k-steps ----
    v2f afrag[16];
#pragma unroll
    for (int ks = 0; ks < 16; ++ks) {
      int mrow = mt * 16 + ll;
      int k = ks * 4 + lh * 2;
      afrag[ks] = *(const v2f*)&hb[mrow * 64 + k];
    }
    for (int ntI = 0; ntI < 8; ++ntI) {
      int nt = ntBase + ntI;
      v8f acc = {0.f, 0.f, 0.f, 0.f, 0.f, 0.f, 0.f, 0.f};
#pragma unroll
      for (int ks = 0; ks < 16; ++ks) {
        int kp = ks * 2 + lh;
        v2f bfrag = wb[kp * 256 + nt * 16 + ll];
        acc = __builtin_amdgcn_wmma_f32_16x16x4_f32(
            false, afrag[ks], false, bfrag, (short)0, acc, false, false);
      }
#pragma unroll
      for (int r8 = 0; r8 < 8; ++r8) {
        gb[(mt * 16 + lh * 8 + r8) * 256 + nt * 16 + ll] = acc[r8];
      }
    }
    __syncthreads();

    // after the conditioning step, merge Whh into the weight buffer
    if (s == 0) {
      for (int idx = t; idx < 32 * 256; idx += 256) {
        int kp = idx >> 8, n = idx & 255;
        v2f v = wb[idx];
        v.x += Whh[n * 64 + 2 * kp];
        v.y += Whh[n * 64 + 2 * kp + 1];
        wb[idx] = v;
      }
    }

    // ---- elementwise phase: c,h update; write hb (+ output for s>=1) -------
    float hv[16];
#pragma unroll
    for (int i = 0; i < 16; ++i) {
      int jj = ec + i;
      float gi = gb[er * 256 + jj]        + bs[jj];
      float gf = gb[er * 256 + 64 + jj]   + bs[64 + jj];
      float gg = gb[er * 256 + 128 + jj]  + bs[128 + jj];
      float go = gb[er * 256 + 192 + jj]  + bs[192 + jj];
      float cn = sigf(gf) * c[i] + sigf(gi) * tanhf(gg);
      c[i] = cn;
      hv[i] = sigf(go) * tanhf(cn);
    }
#pragma unroll
    for (int i = 0; i < 16; ++i) hb[er * 64 + ec + i] = hv[i];
    if (s >= 1) {
      float* op = out + (size_t)(row0 + er) * OUTSTRIDE + (size_t)(s - 1) * 64 + ec;
#pragma unroll
      for (int q = 0; q < 4; ++q) {
        float4 v4 = make_float4(hv[4 * q], hv[4 * q + 1], hv[4 * q + 2], hv[4 * q + 3]);
        ((float4*)op)[q] = v4;
      }
    }
    __syncthreads();
  }
}

extern "C" void kernel_launch(void* const* d_in, const int* in_sizes, int n_in,
                              void* d_out, int out_size, void* d_ws, size_t ws_size,
                              hipStream_t stream) {
  const float* x0    = (const float*)d_in[0];
  const float* x1    = (const float*)d_in[1];
  const float* x2    = (const float*)d_in[2];
  const float* W1    = (const float*)d_in[3];
  const float* b1    = (const float*)d_in[4];
  const float* g1    = (const float*)d_in[5];
  const float* beta1 = (const float*)d_in[6];
  const float* W2    = (const float*)d_in[7];
  const float* b2    = (const float*)d_in[8];
  const float* g2    = (const float*)d_in[9];
  const float* beta2 = (const float*)d_in[10];
  const float* W3    = (const float*)d_in[11];
  const float* b3    = (const float*)d_in[12];
  const float* Wih   = (const float*)d_in[13];
  const float* Whh   = (const float*)d_in[14];
  const float* bih   = (const float*)d_in[15];
  const float* bhh   = (const float*)d_in[16];

  float* ws     = (float*)d_ws;
  float* gstats = ws;                       // 128 used (pad to 256)
  float* a1     = ws + 256;                 // B*32
  float* a2     = a1 + BATCH * 32;          // B*32
  float* mbuf   = a2 + BATCH * 32;          // B*64

  k_zero<<<1, 256, 0, stream>>>(gstats);
  k_mlp1<<<BATCH / 8, 256, 0, stream>>>(x0, x1, x2, W1, b1, a1, gstats);
  k_mlp2<<<BATCH / 256, 256, 0, stream>>>(a1, W2, b2, g1, beta1, a2, gstats);
  k_mlp3<<<BATCH / 256, 256, 0, stream>>>(a2, W3, b3, g2, beta2, gstats, mbuf);
  k_lstm<<<BATCH / 64, 256, 0, stream>>>(mbuf, Wih, Whh, bih, bhh, (float*)d_out);
}